// TrainableAdaptiveKalmanFilter_16518444220660
// MI455X (gfx1250) — compile-verified
//
#include <hip/hip_runtime.h>
#include <hip/hip_bf16.h>

typedef float v2f __attribute__((ext_vector_type(2)));
typedef float v8f __attribute__((ext_vector_type(8)));

// Ap_blk(i,j) for the 16x16 block-diagonal diag(Ap, Ap), Ap = triu(relu(A)), A is 8x8.
__device__ __forceinline__ float apv(const float* __restrict__ A, int i, int j) {
    if ((i >> 3) != (j >> 3)) return 0.0f;   // off-diagonal 8x8 block -> 0
    int ii = i & 7, jj = j & 7;
    if (jj < ii) return 0.0f;                // triu
    float a = A[ii * 8 + jj];
    return a > 0.0f ? a : 0.0f;              // relu
}

// ---------------- x_hat = Ap @ x  (one thread per batch, float4 coalesced) --------------
__global__ __launch_bounds__(256)
void xhat_kernel(const float* __restrict__ x, const float* __restrict__ A,
                 float* __restrict__ out, int Bn) {
    int b = blockIdx.x * blockDim.x + threadIdx.x;
    if (b >= Bn) return;
    const float4* xp = reinterpret_cast<const float4*>(x) + (size_t)b * 2;
    float4 lo = xp[0], hi = xp[1];
    float xv[8] = {lo.x, lo.y, lo.z, lo.w, hi.x, hi.y, hi.z, hi.w};
    float r[8];
#pragma unroll
    for (int i = 0; i < 8; ++i) {
        float acc = 0.0f;
#pragma unroll
        for (int j = 0; j < 8; ++j) {
            if (j >= i) {
                float a = A[i * 8 + j];
                a = a > 0.0f ? a : 0.0f;
                acc = fmaf(a, xv[j], acc);
            }
        }
        r[i] = acc;
    }
    float4 o0 = {r[0], r[1], r[2], r[3]};
    float4 o1 = {r[4], r[5], r[6], r[7]};
    float4* op = reinterpret_cast<float4*>(out) + (size_t)b * 2;
    op[0] = o0;
    op[1] = o1;
}

// ------------- P_hat = Ap P Ap^T + Q ; one wave handles 4 batches via 16x16 tile -------------
__global__ __launch_bounds__(256)
void phat_kernel(const float* __restrict__ x, const float* __restrict__ P,
                 const float* __restrict__ A, const float* __restrict__ sp_p,
                 const float* __restrict__ sv_p, float* __restrict__ out, int Bn) {
    __shared__ float lds[8 * 288];                    // 8 waves * 16 rows * stride 18
    const int  lane    = threadIdx.x & 31;
    const int  wslot   = (threadIdx.x >> 5) * 288;
    const long wid     = (long)blockIdx.x * 8 + (threadIdx.x >> 5);
    const long b0      = wid * 4;
    if (b0 >= Bn) return;
    const int   halfSel = lane >> 4;                  // 0: lanes 0-15, 1: lanes 16-31
    const int   mrow    = lane & 15;
    const float sp      = sp_p[0];
    const float sv      = sv_p[0];
    float*      Pout    = out + (size_t)Bn * 8;

    if (b0 + 4 > Bn) {   // scalar tail fallback (unused when Bn % 4 == 0)
        for (long b = b0; b < Bn; ++b) {
            for (int e = lane; e < 64; e += 32) {
                int r = e >> 3, cc = e & 7;
                float acc2 = 0.0f;
#pragma unroll
                for (int kk = 0; kk < 8; ++kk) {
                    float t = 0.0f;
#pragma unroll
                    for (int j = 0; j < 8; ++j) {
                        if (j >= r) {
                            float a = A[r * 8 + j];
                            a = a > 0.0f ? a : 0.0f;
                            t = fmaf(a, P[b * 64 + j * 8 + kk], t);
                        }
                    }
                    float a2 = (kk >= cc) ? fmaxf(A[cc * 8 + kk], 0.0f) : 0.0f;
                    acc2 = fmaf(t, a2, acc2);
                }
                if (r == cc) {
                    float h = x[b * 8 + 2], wv2 = x[b * 8 + 3];
                    float dv = ((r & 1) ? wv2 : h) * ((r & 4) ? sv : sp);
                    acc2 += dv * dv;
                }
                Pout[b * 64 + e] = acc2;
            }
        }
        return;
    }

    // Constant operand: slices of diag(Ap,Ap). Same registers serve as the
    // A-operand of Ap (step 1) and the B-operand of Ap^T (step 2).
    v2f aop[4];
#pragma unroll
    for (int s = 0; s < 4; ++s) {
        int k0 = 4 * s + 2 * halfSel;
        aop[s].x = apv(A, mrow, k0);
        aop[s].y = apv(A, mrow, k0 + 1);
    }

    // B-operand: Ppack quadrants = batches b0..b0+3, loaded straight into WMMA B layout.
    v2f bop[4];
#pragma unroll
    for (int s = 0; s < 4; ++s) {
#pragma unroll
        for (int t = 0; t < 2; ++t) {
            int  r  = 4 * s + t + 2 * halfSel;        // Ppack row for this VGPR/lane-half
            int  c  = mrow;                           // Ppack col
            long bl = 2 * (r >> 3) + (c >> 3);        // which batch quadrant
            float v = P[(b0 + bl) * 64 + (long)(r & 7) * 8 + (c & 7)];
            if (t == 0) bop[s].x = v; else bop[s].y = v;
        }
    }

    // Step 1: W = diag(Ap,Ap) @ Ppack   (4 accumulating 16x16x4 f32 WMMAs)
    v8f w = {};
#pragma unroll
    for (int s = 0; s < 4; ++s)
        w = __builtin_amdgcn_wmma_f32_16x16x4_f32(false, aop[s], false, bop[s],
                                                  (short)0, w, false, false);

    // Transpose W from C/D layout to A-operand layout through per-wave LDS tile.
#pragma unroll
    for (int v = 0; v < 8; ++v) {
        int M = v + 8 * halfSel;
        lds[wslot + M * 18 + mrow] = w[v];
    }
    asm volatile("s_wait_dscnt 0" ::: "memory");      // wave-local store->load ordering
    v2f wa[4];
#pragma unroll
    for (int s = 0; s < 4; ++s) {
        int cb = 4 * s + 2 * halfSel;
        wa[s].x = lds[wslot + mrow * 18 + cb];
        wa[s].y = lds[wslot + mrow * 18 + cb + 1];
    }

    // Q on the quadrant diagonals, preloaded as the C operand.
    int   bl = 2 * halfSel + ((lane >> 3) & 1);       // this lane's batch quadrant
    float h  = x[(b0 + bl) * 8 + 2];
    float wv = x[(b0 + bl) * 8 + 3];
    v8f acc;
#pragma unroll
    for (int v = 0; v < 8; ++v) {
        float dv = ((v & 1) ? wv : h) * ((v & 4) ? sv : sp);
        acc[v] = ((lane & 7) == v) ? dv * dv : 0.0f;
    }

    // Step 2: P_hat_pack = W @ diag(Ap^T,Ap^T) + Q
#pragma unroll
    for (int s = 0; s < 4; ++s)
        acc = __builtin_amdgcn_wmma_f32_16x16x4_f32(false, wa[s], false, aop[s],
                                                    (short)0, acc, false, false);

    // Store: C/D element (M=v+8*halfSel, N=lane&15) -> batch bl, row v, col lane&7.
    int c = lane & 7;
#pragma unroll
    for (int v = 0; v < 8; ++v)
        Pout[(b0 + bl) * 64 + (long)v * 8 + c] = acc[v];
}

extern "C" void kernel_launch(void* const* d_in, const int* in_sizes, int n_in,
                              void* d_out, int out_size, void* d_ws, size_t ws_size,
                              hipStream_t stream) {
    const float* x  = (const float*)d_in[0];
    const float* P  = (const float*)d_in[1];
    const float* A  = (const float*)d_in[2];
    const float* sp = (const float*)d_in[3];
    const float* sv = (const float*)d_in[4];
    float* out = (float*)d_out;
    int Bn = in_sizes[0] / 8;

    int blocksX = (Bn + 255) / 256;
    xhat_kernel<<<blocksX, 256, 0, stream>>>(x, A, out, Bn);

    int nwaves  = (Bn + 3) / 4;        // one wave per 4 batches
    int blocksP = (nwaves + 7) / 8;    // 8 waves per 256-thread block
    phat_kernel<<<blocksP, 256, 0, stream>>>(x, P, A, sp, sv, out, Bn);
}